// MultiHeadAttention_27522150432976
// MI455X (gfx1250) — compile-verified
//
#include <hip/hip_runtime.h>

typedef __attribute__((ext_vector_type(2))) float v2f;
typedef __attribute__((ext_vector_type(8))) float v8f;

#define WT_STRIDE 66   // 64 + 2 padding floats to spread LDS banks

// ---------------------------------------------------------------------------
// Kernel 1: fused node projections  Q_c, Q_p, K, V  via V_WMMA_F32_16X16X4_F32
// Each wave: 16 rows x 64 cols for all 4 outputs. 8 waves / block -> 128 rows.
// ---------------------------------------------------------------------------
__global__ __launch_bounds__(256)
void proj_wmma_kernel(const float* __restrict__ h_c,
                      const float* __restrict__ h_p,
                      const float* __restrict__ Wqc, const float* __restrict__ bqc,
                      const float* __restrict__ Wqp, const float* __restrict__ bqp,
                      const float* __restrict__ Wk,  const float* __restrict__ bk,
                      const float* __restrict__ Wv,  const float* __restrict__ bv,
                      float* __restrict__ Qc, float* __restrict__ Qp,
                      float* __restrict__ Kn, float* __restrict__ Vn,
                      int n)
{
    __shared__ float sWT[4][64 * WT_STRIDE];   // transposed weights: [col][k]
    __shared__ float sB[4][64];

    const int tid = threadIdx.x;
    const float* Ws[4] = {Wqc, Wqp, Wk, Wv};
    const float* Bs[4] = {bqc, bqp, bk, bv};
    for (int o = 0; o < 4; ++o) {
        for (int i = tid; i < 4096; i += 256) {
            int col = i >> 6, k = i & 63;
            sWT[o][col * WT_STRIDE + k] = Ws[o][k * 64 + col];
        }
        if (tid < 64) sB[o][tid] = Bs[o][tid];
    }
    __syncthreads();

    const int wave = tid >> 5;
    const int lane = tid & 31;
    const int half = lane >> 4;      // 0: K pair {0,1}/rows M,  1: K pair {2,3}/rows M+8
    const int lrow = lane & 15;      // A-matrix row within tile
    const int ncol = lane & 15;      // B/C column within tile

    const int m0 = (blockIdx.x * 8 + wave) * 16;
    if (m0 >= n) return;             // N % 16 == 0 -> whole-wave uniform exit

    const int row = m0 + lrow;

    // Preload A fragments for both activation matrices (16 k-steps x v2f each)
    v2f a_c[16], a_p[16];
#pragma unroll
    for (int ks = 0; ks < 16; ++ks) {
        int k = ks * 4 + 2 * half;
        a_c[ks] = *(const v2f*)(h_c + row * 64 + k);
        a_p[ks] = *(const v2f*)(h_p + row * 64 + k);
    }

    float* outs[4] = {Qc, Qp, Kn, Vn};

#pragma unroll
    for (int o = 0; o < 4; ++o) {
        float* op = outs[o];
#pragma unroll
        for (int nt = 0; nt < 4; ++nt) {
            const int ncg = nt * 16 + ncol;
            v8f acc = {0.f, 0.f, 0.f, 0.f, 0.f, 0.f, 0.f, 0.f};
#pragma unroll
            for (int ks = 0; ks < 16; ++ks) {
                int k = ks * 4 + 2 * half;
                v2f b = *(const v2f*)(&sWT[o][ncg * WT_STRIDE + k]);
                v2f a = (o == 1) ? a_p[ks] : a_c[ks];
                acc = __builtin_amdgcn_wmma_f32_16x16x4_f32(
                    /*neg_a=*/false, a, /*neg_b=*/false, b,
                    /*c_mod=*/(short)0, acc, /*reuse_a=*/false, /*reuse_b=*/false);
            }
            const float bias = sB[o][ncg];
#pragma unroll
            for (int r = 0; r < 8; ++r) {       // C layout: VGPR r -> M = r + 8*half
                int m = m0 + r + 8 * half;
                op[m * 64 + ncg] = acc[r] + bias;
            }
        }
    }
}

// ---------------------------------------------------------------------------
// Kernel 2: zero outputs
// ---------------------------------------------------------------------------
__global__ void zero_kernel(float* __restrict__ p, int n)
{
    int i = blockIdx.x * blockDim.x + threadIdx.x;
    if (i < n) p[i] = 0.0f;
}

// ---------------------------------------------------------------------------
// Kernel 3: per-(edge, head) attention + atomic scatter-add.
// All node arrays (~150 MB) fit in the 192 MB L2, so gathers/atomics stay
// on-chip.
// ---------------------------------------------------------------------------
__global__ __launch_bounds__(256)
void edge_attn_kernel(const int* __restrict__ src, const int* __restrict__ dst,
                      const float* __restrict__ Qc, const float* __restrict__ Qp,
                      const float* __restrict__ Kn, const float* __restrict__ Vn,
                      float* __restrict__ out_c, float* __restrict__ out_p,
                      int total /* = E * 8 heads */)
{
    int idx = blockIdx.x * blockDim.x + threadIdx.x;
    if (idx >= total) return;
    const int e = idx >> 3;
    const int h = idx & 7;
    const int s = src[e];
    const int d = dst[e];
    const int bs = s * 64 + h * 8;
    const int bd = d * 64 + h * 8;

    const float4 k0  = *(const float4*)(Kn + bs);
    const float4 k1  = *(const float4*)(Kn + bs + 4);
    const float4 v0  = *(const float4*)(Vn + bs);
    const float4 v1  = *(const float4*)(Vn + bs + 4);
    const float4 qc0 = *(const float4*)(Qc + bd);
    const float4 qc1 = *(const float4*)(Qc + bd + 4);
    const float4 qp0 = *(const float4*)(Qp + bd);
    const float4 qp1 = *(const float4*)(Qp + bd + 4);

    float dc = qc0.x*k0.x + qc0.y*k0.y + qc0.z*k0.z + qc0.w*k0.w
             + qc1.x*k1.x + qc1.y*k1.y + qc1.z*k1.z + qc1.w*k1.w;
    float dp = qp0.x*k0.x + qp0.y*k0.y + qp0.z*k0.z + qp0.w*k0.w
             + qp1.x*k1.x + qp1.y*k1.y + qp1.z*k1.z + qp1.w*k1.w;

    const float sc = __expf(fminf(fmaxf(dc * 0.125f, -5.0f), 5.0f));  // SCALE = 1/sqrt(64)
    const float sp = __expf(fminf(fmaxf(dp * 0.125f, -5.0f), 5.0f));

    atomicAdd(out_c + bd + 0, v0.x * sc);
    atomicAdd(out_c + bd + 1, v0.y * sc);
    atomicAdd(out_c + bd + 2, v0.z * sc);
    atomicAdd(out_c + bd + 3, v0.w * sc);
    atomicAdd(out_c + bd + 4, v1.x * sc);
    atomicAdd(out_c + bd + 5, v1.y * sc);
    atomicAdd(out_c + bd + 6, v1.z * sc);
    atomicAdd(out_c + bd + 7, v1.w * sc);

    atomicAdd(out_p + bd + 0, v0.x * sp);
    atomicAdd(out_p + bd + 1, v0.y * sp);
    atomicAdd(out_p + bd + 2, v0.z * sp);
    atomicAdd(out_p + bd + 3, v0.w * sp);
    atomicAdd(out_p + bd + 4, v1.x * sp);
    atomicAdd(out_p + bd + 5, v1.y * sp);
    atomicAdd(out_p + bd + 6, v1.z * sp);
    atomicAdd(out_p + bd + 7, v1.w * sp);
}

// ---------------------------------------------------------------------------
extern "C" void kernel_launch(void* const* d_in, const int* in_sizes, int n_in,
                              void* d_out, int out_size, void* d_ws, size_t ws_size,
                              hipStream_t stream)
{
    const float* h_c  = (const float*)d_in[0];
    const float* h_p  = (const float*)d_in[1];
    const int*   src  = (const int*)  d_in[2];
    const int*   dst  = (const int*)  d_in[3];
    const float* Wqcw = (const float*)d_in[4];
    const float* Wqcb = (const float*)d_in[5];
    const float* Wqpw = (const float*)d_in[6];
    const float* Wqpb = (const float*)d_in[7];
    const float* Wkw  = (const float*)d_in[8];
    const float* Wkb  = (const float*)d_in[9];
    const float* Wvw  = (const float*)d_in[10];
    const float* Wvb  = (const float*)d_in[11];

    const int n = in_sizes[0] / 64;   // N nodes
    const int e = in_sizes[2];        // E edges

    float* Qc = (float*)d_ws;         // 4 * n * 64 floats of scratch
    float* Qp = Qc + (size_t)n * 64;
    float* Kn = Qp + (size_t)n * 64;
    float* Vn = Kn + (size_t)n * 64;

    float* out_c = (float*)d_out;
    float* out_p = out_c + (size_t)n * 64;

    // 1) zero the scatter targets
    {
        int total = 2 * n * 64;
        zero_kernel<<<(total + 255) / 256, 256, 0, stream>>>((float*)d_out, total);
    }

    // 2) WMMA projections (128 rows per block)
    {
        int blocks = (n + 127) / 128;
        proj_wmma_kernel<<<blocks, 256, 0, stream>>>(
            h_c, h_p, Wqcw, Wqcb, Wqpw, Wqpb, Wkw, Wkb, Wvw, Wvb,
            Qc, Qp, Kn, Vn, n);
    }

    // 3) edge attention + scatter
    {
        int total = e * 8;
        edge_attn_kernel<<<(total + 255) / 256, 256, 0, stream>>>(
            src, dst, Qc, Qp, Kn, Vn, out_c, out_p, total);
    }
}